// CenterNet_68753836474735
// MI455X (gfx1250) — compile-verified
//
#include <hip/hip_runtime.h>

// ---------------- problem constants (from reference) ----------------
#define MM   349184
#define CC   80
#define MC   (MM * CC)            // 27,934,720
#define MC4  (MC / 4)             // 6,983,680 float4s
#define NN   256
#define LLn  5
#define KKn  9
#define NL   (NN * LLn)           // 1280
#define NLK  (NN * LLn * KKn)     // 11520 == 45 * 256 exactly

#define NEG_BLOCKS 4096
#define LOC_BLOCKS (NLK / 256)    // 45

// workspace layout (floats)
#define WS_POS  NEG_BLOCKS
#define WS_LOC  (NEG_BLOCKS + 1)
#define WS_W    (WS_LOC + LOC_BLOCKS)

typedef __attribute__((ext_vector_type(2))) float v2f;
typedef __attribute__((ext_vector_type(4))) float v4f;
typedef __attribute__((ext_vector_type(8))) float v8f;

// ---------------- wave32 sum via V_WMMA_F32_16X16X4_F32 ----------------
// A = per-lane accumulator in A[0], 0 in A[1]; B = all-ones.
// D[m][n] = sum_k A[m][k]  (independent of n), so sum over one column of D
// equals the sum of every A element = sum over all 32 lanes of x.
// Column j lives in lanes j (rows 0-7) and j+16 (rows 8-15): add the 8 D
// regs locally, then one shfl_xor(16). Requires EXEC all-ones (true at all
// call sites: full blocks, post-reconvergence).
__device__ __forceinline__ float wave_sum(float x) {
  v2f a; a[0] = x;    a[1] = 0.0f;
  v2f b; b[0] = 1.0f; b[1] = 1.0f;
  v8f c = {};
  c = __builtin_amdgcn_wmma_f32_16x16x4_f32(false, a, false, b,
                                            (short)0, c, false, false);
  float s = ((c[0] + c[1]) + (c[2] + c[3])) + ((c[4] + c[5]) + (c[6] + c[7]));
  s += __shfl_xor(s, 16, 32);
  return s;
}

// block (256 threads = 8 waves) sum; result valid on wave 0
__device__ __forceinline__ float block_sum(float x) {
  __shared__ float lds[8];
  float w = wave_sum(x);
  const int lane = threadIdx.x & 31;
  const int wid  = threadIdx.x >> 5;
  if (lane == 0) lds[wid] = w;
  __syncthreads();
  float r = (threadIdx.x < (blockDim.x >> 5)) ? lds[threadIdx.x] : 0.0f;
  if (wid == 0) r = wave_sum(r);
  __syncthreads();          // allow reuse of lds by a subsequent call
  return r;
}

__device__ __forceinline__ float sigf(float x) {
  return __builtin_amdgcn_rcpf(1.0f + __expf(-x));
}

// ---------------- kernel 1: negative focal loss (bandwidth bound) ----------------
__global__ void __launch_bounds__(256)
neg_kernel(const v4f* __restrict__ lg, const v4f* __restrict__ hm,
           float* __restrict__ part) {
  const int stride = gridDim.x * blockDim.x;
  float acc = 0.0f;
  for (int i = blockIdx.x * blockDim.x + threadIdx.x; i < MC4; i += stride) {
    v4f x4 = __builtin_nontemporal_load(lg + i);   // global_load_b128, TH=NT
    v4f t4 = __builtin_nontemporal_load(hm + i);
#pragma unroll
    for (int c = 0; c < 4; ++c) {
      float p = sigf(x4[c]);
      if (p < 0.85f) {                       // IGNORE_HIGH_FP
        float om  = 1.0f - t4[c];            // neg weight (1-hm)^4
        float om2 = om * om;
        float p2  = p * p;
        acc += -__logf(1.0f - p) * (p2 * p2) * (om2 * om2);
      }
    }
  }
  float s = block_sum(acc);
  if (threadIdx.x == 0) part[blockIdx.x] = s;
}

// ---------------- kernel 2: positive focal loss (1280 gathers) ----------------
__global__ void __launch_bounds__(256)
pos_kernel(const float* __restrict__ lg, const int* __restrict__ cinds,
           const int* __restrict__ labels, const int* __restrict__ masks,
           float* __restrict__ ws) {
  float acc = 0.0f;
  for (int i = threadIdx.x; i < NL; i += 256) {
    int n   = i / LLn;
    int idx = cinds[i];
    int lab = labels[n];
    float p = sigf(lg[idx * CC + lab]);
    p = fminf(fmaxf(p, 1e-4f), 1.0f - 1e-4f);   // SIGMOID_CLAMP
    float om  = 1.0f - p;
    float om2 = om * om;
    float e   = -__logf(p) * (om2 * om2);
    if (masks[i]) acc += e;
  }
  float s = block_sum(acc);
  if (threadIdx.x == 0) ws[WS_POS] = s;
}

// ---------------- kernel 3: GIoU loss (11520 entries, 1 thread each) ----------------
__global__ void __launch_bounds__(256)
loc_kernel(const float* __restrict__ lg, const v4f* __restrict__ rp4,
           const v4f* __restrict__ c33r4, const int* __restrict__ masks,
           const int* __restrict__ c33i, const int* __restrict__ c33m,
           float* __restrict__ loc_part, float* __restrict__ w_part) {
  const int e  = blockIdx.x * 256 + threadIdx.x;     // grid covers NLK exactly
  const int nl = e / KKn;
  const bool m33 = (c33m[e] != 0) && (masks[nl] != 0);
  float term = 0.0f, w = 0.0f;
  if (m33) {
    int ind = c33i[e];
    // max_c sigmoid(row) == sigmoid(max_c row): 20 x b128 row loads
    const v4f* row = (const v4f*)(lg + ind * CC);
    float mx = -3.4e38f;
#pragma unroll
    for (int j = 0; j < CC / 4; ++j) {
      v4f v = row[j];
      mx = fmaxf(mx, fmaxf(fmaxf(v[0], v[1]), fmaxf(v[2], v[3])));
    }
    w = sigf(mx);

    v4f pr = rp4[ind];
    v4f tg = c33r4[e];
    float pl = pr[0], pt = pr[1], pe = pr[2], pb = pr[3];
    float tl = tg[0], tt = tg[1], tr = tg[2], tb = tg[3];
    float ta = (tl + tr) * (tt + tb);
    float pa = (pl + pe) * (pt + pb);
    float wi = fminf(pl, tl) + fminf(pe, tr);
    float gw = fmaxf(pl, tl) + fmaxf(pe, tr);
    float hi = fminf(pb, tb) + fminf(pt, tt);
    float gh = fmaxf(pb, tb) + fmaxf(pt, tt);
    float ac  = gw * gh + 1e-7f;
    float ai  = wi * hi;
    float au  = ta + pa - ai;
    float iou = (ai + 1.0f) / (au + 1.0f);
    float gio = iou - (ac - au) / ac;
    term = (1.0f - gio) * w;
  }
  float ls   = block_sum(term);
  float wsum = block_sum(w);
  if (threadIdx.x == 0) { loc_part[blockIdx.x] = ls; w_part[blockIdx.x] = wsum; }
}

// ---------------- kernel 4: finalize ----------------
__global__ void __launch_bounds__(256)
fin_kernel(const float* __restrict__ ws, float* __restrict__ out) {
  float negs = 0.0f;
  for (int i = threadIdx.x; i < NEG_BLOCKS; i += 256) negs += ws[i];
  negs = block_sum(negs);

  float locs = 0.0f, wsum = 0.0f;
  if (threadIdx.x < LOC_BLOCKS) {
    locs = ws[WS_LOC + threadIdx.x];
    wsum = ws[WS_W + threadIdx.x];
  }
  locs = block_sum(locs);
  wsum = block_sum(wsum);

  if (threadIdx.x == 0) {
    float pos = ws[WS_POS];
    const float num_pos_avg = 256.0f;                 // max(N,1)
    out[0] = 0.5f * 0.25f * pos  / num_pos_avg;       // POS_WEIGHT*ALPHA
    out[1] = 0.5f * 0.75f * negs / num_pos_avg;       // NEG_WEIGHT*(1-ALPHA)
    out[2] = 2.0f * locs / fmaxf(wsum, 1.0f);         // REG_WEIGHT / reg_norm
  }
}

extern "C" void kernel_launch(void* const* d_in, const int* in_sizes, int n_in,
                              void* d_out, int out_size, void* d_ws, size_t ws_size,
                              hipStream_t stream) {
  (void)in_sizes; (void)n_in; (void)out_size; (void)ws_size;
  const float* lg    = (const float*)d_in[0];   // logits_pred   (M,C)
  const v4f*   rp4   = (const v4f*)  d_in[1];   // reg_pred      (M,4)
  const float* hm    = (const float*)d_in[2];   // flattened_hms (M,C)
  const v4f*   c33r4 = (const v4f*)  d_in[3];   // c33_regs      (N,L,K,4)
  const int*   cinds = (const int*)  d_in[4];   // center_inds   (N,L)
  const int*   labels= (const int*)  d_in[5];   // labels        (N,)
  const int*   masks = (const int*)  d_in[6];   // masks         (N,L)
  const int*   c33i  = (const int*)  d_in[7];   // c33_inds      (N,L,K)
  const int*   c33m  = (const int*)  d_in[8];   // c33_mask      (N,L,K)
  float* ws  = (float*)d_ws;
  float* out = (float*)d_out;

  neg_kernel<<<NEG_BLOCKS, 256, 0, stream>>>((const v4f*)lg, (const v4f*)hm, ws);
  pos_kernel<<<1, 256, 0, stream>>>(lg, cinds, labels, masks, ws);
  loc_kernel<<<LOC_BLOCKS, 256, 0, stream>>>(lg, rp4, c33r4, masks, c33i, c33m,
                                             ws + WS_LOC, ws + WS_W);
  fin_kernel<<<1, 256, 0, stream>>>(ws, out);
}